// GCC_61649960566859
// MI455X (gfx1250) — compile-verified
//
#include <hip/hip_runtime.h>

#define B_DIM   64
#define N_CH    12
#define K_LEN   4096
#define KF      2049          // rfft bins
#define TAU_MAX 25
#define NLAG    51            // 2*TAU_MAX+1
#define P_DIM   144           // N_CH*N_CH (== 9*16, no row padding needed)
#define KQ_PAD  2050          // KF padded so q = 2t+half covers the tail step
#define COL_PAD 64            // 51 lags padded to 4 WMMA col tiles

#define PI_F 3.14159265358979323846f

typedef __attribute__((ext_vector_type(2))) float v2f;
typedef __attribute__((ext_vector_type(4))) float v4f;
typedef __attribute__((ext_vector_type(8))) float v8f;

// ---------------------------------------------------------------------------
// Kernel 1: 4096-pt complex FFT (radix-2 Stockham, auto-sort) in LDS + PHAT.
// One workgroup per (b,n) signal; 256 threads = 8 wave32.
// Output: spec[sig][k] = (re, im) of xf[k]/(|xf[k]|+eps), k = 0..2048.
// ---------------------------------------------------------------------------
__global__ __launch_bounds__(256) void fft_phat_kernel(const float* __restrict__ x,
                                                       float* __restrict__ spec) {
  __shared__ float bufRe[2][K_LEN];   // 2 x 16 KB
  __shared__ float bufIm[2][K_LEN];   // 2 x 16 KB  (64 KB total)
  const int sig = blockIdx.x;         // b*12 + n, 0..767
  const int tid = threadIdx.x;

  const float* xs = x + (size_t)sig * K_LEN;
  for (int i = tid; i < K_LEN; i += 256) {
    bufRe[0][i] = xs[i];
    bufIm[0][i] = 0.0f;
  }

  int cur = 0;
  for (int Ns = 1; Ns < K_LEN; Ns <<= 1) {
    __syncthreads();
    const float* sr = bufRe[cur];
    const float* si = bufIm[cur];
    float* dr = bufRe[cur ^ 1];
    float* di = bufIm[cur ^ 1];
    const float angScale = -PI_F / (float)Ns;
    for (int j = tid; j < K_LEN / 2; j += 256) {
      const int jm = j & (Ns - 1);
      const float vr0 = sr[j],             vi0 = si[j];
      const float vr1 = sr[j + K_LEN / 2], vi1 = si[j + K_LEN / 2];
      float s, c;
      __sincosf(angScale * (float)jm, &s, &c);      // w = exp(-2*pi*i*jm/(2*Ns))
      const float tr = c * vr1 - s * vi1;
      const float ti = c * vi1 + s * vr1;
      const int d = (j << 1) - jm;                  // (j/Ns)*2Ns + jm
      dr[d]      = vr0 + tr;  di[d]      = vi0 + ti;
      dr[d + Ns] = vr0 - tr;  di[d + Ns] = vi0 - ti;
    }
    cur ^= 1;
  }
  __syncthreads();

  // PHAT normalize bins 0..2048 and write interleaved (re, im).
  float* outp = spec + (size_t)sig * (2 * KF);
  for (int k = tid; k < KF; k += 256) {
    const float a = bufRe[cur][k];
    const float b = bufIm[cur][k];
    const float inv = 1.0f / (sqrtf(a * a + b * b) + 1e-12f);
    outp[2 * k]     = a * inv;
    outp[2 * k + 1] = b * inv;
  }
}

// ---------------------------------------------------------------------------
// Kernel 2: irfft->lag basis, pair-interleaved layout keyed by frequency q:
//   W3[q][j][ct] = ( w_q*cos(2*pi*q*tau/K), w_q*sin(2*pi*q*tau/K) ),
//   col = ct*16 + j,  flat float index = ((q*16 + j)*4 + ct)*2 + h.
// Lane j in kernel 3 loads 32 contiguous bytes per frequency: two b128s whose
// component pairs {0,1},{2,3} ARE the B fragments (basis rows 2q, 2q+1) for
// col tiles (0,1) and (2,3) -- adjacent VGPR pairs straight from the load.
// Rows q >= 2049 and cols >= 51 are zero padding.
// ---------------------------------------------------------------------------
__global__ __launch_bounds__(256) void basis_kernel(float* __restrict__ W3) {
  const int idx = blockIdx.x * 256 + threadIdx.x;
  if (idx >= KQ_PAD * COL_PAD * 2) return;
  const int h  = idx & 1;           // 0 = cos row (2q), 1 = sin row (2q+1)
  const int ct = (idx >> 1) & 3;    // column tile
  const int j  = (idx >> 3) & 15;   // lane index within tile
  const int q  = idx >> 7;          // frequency bin
  const int col = ct * 16 + j;
  float v = 0.0f;
  if (col < NLAG && q < KF) {
    const int tau = (col <= TAU_MAX) ? col : col - NLAG;   // negative lags wrap mod K
    const int mm  = (q * tau) & (K_LEN - 1);               // exact range reduction
    const float ang = (float)mm * (2.0f * PI_F / (float)K_LEN);
    const float w = (q == 0 || q == KF - 1) ? (1.0f / (float)K_LEN)
                                            : (2.0f / (float)K_LEN);
    float s, c;
    __sincosf(ang, &s, &c);
    v = h ? (w * s) : (w * c);      // sin row pairs with -Im(P) in the A frag
  }
  W3[idx] = v;
}

// ---------------------------------------------------------------------------
// Kernel 3: G[b, p=(n,m), tau] = M[b,p,:] @ W[:,tau] via V_WMMA_F32_16X16X4_F32.
// Grid: (3 row-tile groups, 64 batches); block = 96 threads = 3 waves; each
// wave owns one 16-row tile of the 144 pair rows and ALL 4 column tiles
// (4 accumulators), so the A fragment (Re(P), -Im(P)) built from the PHAT
// spectra is amortized over 4 WMMAs; the [B,144,4098] pair tensor is never
// materialized. Per k-step: 2x global_load_b64 (spec) + 2x global_load_b128
// (basis) feed 4 WMMAs.
// A layout (16x4 f32): lanes 0-15 M=lane, lanes 16-31 M=lane-16;
//                      VGPR0 = K0/K2 (lane halves), VGPR1 = K1/K3.
// B layout (4x16 f32): lanes hold N; VGPR0 = K0/K2, VGPR1 = K1/K3.
// C/D layout: VGPR v -> M = v (lanes 0-15) / v+8 (lanes 16-31), N = lane&15.
// ---------------------------------------------------------------------------
__global__ __launch_bounds__(96) void gcc_wmma_kernel(const float* __restrict__ spec,
                                                      const float* __restrict__ W3,
                                                      float* __restrict__ out) {
  const int b    = blockIdx.y;                      // 0..63 batch
  const int wave = threadIdx.x >> 5;                // 0..2
  const int rt   = blockIdx.x * 3 + wave;           // 0..8 row tile
  const int lane = threadIdx.x & 31;
  const int half = lane >> 4;                       // K lane-half per WMMA layout
  const int lidx = lane & 15;

  const int p = rt * 16 + lidx;                     // pair row 0..143 = n*12 + m
  const int n = p / N_CH;
  const int m = p % N_CH;

  const float2* zn = (const float2*)spec + (size_t)(b * N_CH + n) * KF;
  const float2* zm = (const float2*)spec + (size_t)(b * N_CH + m) * KF;
  // lane base: j*2 v4f; per-frequency stride: 8 v4f (= 16 float2 pairs)
  const v4f* w3 = (const v4f*)W3 + (size_t)lidx * 2;

  v8f acc0 = {}, acc1 = {}, acc2 = {}, acc3 = {};

  // Main loop: q = 2t + half <= 2047 always valid -> no clamps, pure
  // pointer-bump + immediate-offset addressing.
#pragma unroll 2
  for (int t = 0; t < 1024; ++t) {
    const int q = 2 * t + half;
    const float2 an = zn[q];
    const float2 am = zm[q];
    v2f A;
    A.x = an.x * am.x + an.y * am.y;                //  Re(z_n conj(z_m))
    A.y = an.x * am.y - an.y * am.x;                // -Im(z_n conj(z_m))

    const v4f wa = w3[(size_t)q * 8];               // (cos,sin) pairs, tiles 0,1
    const v4f wb = w3[(size_t)q * 8 + 1];           // (cos,sin) pairs, tiles 2,3

    v2f Bf;
    Bf.x = wa[0]; Bf.y = wa[1];
    acc0 = __builtin_amdgcn_wmma_f32_16x16x4_f32(false, A, false, Bf, (short)0, acc0, false, false);
    Bf.x = wa[2]; Bf.y = wa[3];
    acc1 = __builtin_amdgcn_wmma_f32_16x16x4_f32(false, A, false, Bf, (short)0, acc1, false, false);
    Bf.x = wb[0]; Bf.y = wb[1];
    acc2 = __builtin_amdgcn_wmma_f32_16x16x4_f32(false, A, false, Bf, (short)0, acc2, false, false);
    Bf.x = wb[2]; Bf.y = wb[3];
    acc3 = __builtin_amdgcn_wmma_f32_16x16x4_f32(false, A, false, Bf, (short)0, acc3, false, false);
  }

  // Tail step t = 1024: q = 2048 (half=0) is the Nyquist bin, q = 2049
  // (half=1) is padding -> mask by multiply (EXEC stays all-ones for WMMA;
  // basis row q=2049 is zero anyway).
  {
    const int q  = 2048 + half;
    const int qc = (q <= KF - 1) ? q : (KF - 1);
    const float msk = (q <= KF - 1) ? 1.0f : 0.0f;
    const float2 an = zn[qc];
    const float2 am = zm[qc];
    v2f A;
    A.x = msk * (an.x * am.x + an.y * am.y);
    A.y = msk * (an.x * am.y - an.y * am.x);

    const v4f wa = w3[(size_t)q * 8];
    const v4f wb = w3[(size_t)q * 8 + 1];

    v2f Bf;
    Bf.x = wa[0]; Bf.y = wa[1];
    acc0 = __builtin_amdgcn_wmma_f32_16x16x4_f32(false, A, false, Bf, (short)0, acc0, false, false);
    Bf.x = wa[2]; Bf.y = wa[3];
    acc1 = __builtin_amdgcn_wmma_f32_16x16x4_f32(false, A, false, Bf, (short)0, acc1, false, false);
    Bf.x = wb[0]; Bf.y = wb[1];
    acc2 = __builtin_amdgcn_wmma_f32_16x16x4_f32(false, A, false, Bf, (short)0, acc2, false, false);
    Bf.x = wb[2]; Bf.y = wb[3];
    acc3 = __builtin_amdgcn_wmma_f32_16x16x4_f32(false, A, false, Bf, (short)0, acc3, false, false);
  }

  // Store D tiles: VGPR v -> row half*8+v of this row tile, col ct*16+lidx.
  const v8f accs[4] = {acc0, acc1, acc2, acc3};
#pragma unroll
  for (int ct = 0; ct < 4; ++ct) {
    const int col = ct * 16 + lidx;
    if (col < NLAG) {
#pragma unroll
      for (int v = 0; v < 8; ++v) {
        const int pp = rt * 16 + half * 8 + v;
        out[((size_t)b * P_DIM + pp) * NLAG + col] = accs[ct][v];
      }
    }
  }
}

// ---------------------------------------------------------------------------
extern "C" void kernel_launch(void* const* d_in, const int* in_sizes, int n_in,
                              void* d_out, int out_size, void* d_ws, size_t ws_size,
                              hipStream_t stream) {
  const float* x = (const float*)d_in[0];   // [64, 12, 4096] f32
  float* out = (float*)d_out;               // [64, 12, 12, 51] f32

  // Workspace layout: PHAT spectra then pair-interleaved basis (~13.6 MB).
  float* spec = (float*)d_ws;                                   // 768*2049*2 f32
  float* W3   = spec + (size_t)B_DIM * N_CH * KF * 2;           // 2050*128 f32

  fft_phat_kernel<<<B_DIM * N_CH, 256, 0, stream>>>(x, spec);
  basis_kernel<<<(KQ_PAD * COL_PAD * 2 + 255) / 256, 256, 0, stream>>>(W3);
  gcc_wmma_kernel<<<dim3(3, B_DIM), 96, 0, stream>>>(spec, W3, out);
}